// QNet_59906203844854
// MI455X (gfx1250) — compile-verified
//
#include <hip/hip_runtime.h>
#include <math.h>

// ---------------- problem constants (match reference) ----------------
#define NQ      12
#define DIM     4096           // 2^NQ
#define LREP    4
#define B_BATCH 2048
#define T_STEPS 50
#define TIN     300
#define H       12
#define AIN     74
#define VIN     35
#define PI_F    3.14159265358979323846f

typedef __attribute__((ext_vector_type(2))) float v2f;
typedef __attribute__((ext_vector_type(8))) float v8f;

// =====================================================================
// Kernel 1: a_p / v_p projections (pre-sigmoid), z_av[B,24]
// =====================================================================
__global__ void proj_av_kernel(const float* __restrict__ x_a, const float* __restrict__ x_v,
                               const float* __restrict__ W_a, const float* __restrict__ b_a,
                               const float* __restrict__ W_v, const float* __restrict__ b_v,
                               float* __restrict__ z_av) {
    int idx = blockIdx.x * blockDim.x + threadIdx.x;     // [B*24]
    if (idx >= B_BATCH * 24) return;
    int b = idx / 24, j = idx % 24;
    float acc;
    if (j < H) {
        acc = b_a[j];
        const float* xr = x_a + (long long)b * AIN;
        const float* wr = W_a + j * AIN;
        for (int k = 0; k < AIN; ++k) acc += xr[k] * wr[k];
    } else {
        int jj = j - H;
        acc = b_v[jj];
        const float* xr = x_v + (long long)b * VIN;
        const float* wr = W_v + jj * VIN;
        for (int k = 0; k < VIN; ++k) acc += xr[k] * wr[k];
    }
    z_av[idx] = acc;
}

// =====================================================================
// Kernel 2: G[B*T, H] = x_t @ W_ih^T + b_ih via V_WMMA_F32_16X16X4_F32
//   M = 102400, K = 300 (75 chunks of 4), N = 12 padded to 16.
//   4 waves per block, one 16-row tile per wave. W_ih^T staged in LDS.
// =====================================================================
#define GEMM_WAVES 4
__global__ void __launch_bounds__(32 * GEMM_WAVES, 1)
rnn_in_gemm_kernel(const float* __restrict__ x_t, const float* __restrict__ W_ih,
                   const float* __restrict__ b_ih, float* __restrict__ G) {
    __shared__ float Bs[TIN * 16];     // Bs[k*16 + n] = W_ih^T[k][n], N padded to 16
    __shared__ float bih_s[16];

    const int tid = threadIdx.x;
    for (int i = tid; i < TIN * 16; i += blockDim.x) {
        int k = i >> 4, n = i & 15;
        Bs[i] = (n < H) ? W_ih[n * TIN + k] : 0.0f;
    }
    if (tid < 16) bih_s[tid] = (tid < H) ? b_ih[tid] : 0.0f;
    __syncthreads();

    const int wave  = tid >> 5;
    const int lane  = tid & 31;
    const long long m0 = ((long long)blockIdx.x * GEMM_WAVES + wave) * 16;
    const int mrow  = lane & 15;            // A: M index / D: N index
    const int khalf = (lane >> 4) << 1;     // lanes 16-31 carry K = k0+2, k0+3

    const float* arow = x_t + (m0 + mrow) * TIN;

    v8f c = {};
    for (int k0 = 0; k0 < TIN; k0 += 4) {
        const int kk = k0 + khalf;
        v2f a = *(const v2f*)(arow + kk);          // 8B-aligned: row stride 1200B, kk even
        v2f bf;
        bf.x = Bs[kk * 16 + mrow];
        bf.y = Bs[(kk + 1) * 16 + mrow];
        if (k0 + 4 < TIN) __builtin_prefetch(arow + kk + 4, 0, 1);
        // D = A(16x4,f32) * B(4x16,f32) + C ; args: neg_a,A,neg_b,B,c_mod,C,reuse_a,reuse_b
        c = __builtin_amdgcn_wmma_f32_16x16x4_f32(false, a, false, bf, (short)0, c,
                                                  false, false);
    }

    // D layout: VGPR j -> row (lane<16 ? j : j+8), col = lane&15
    const int n = lane & 15;
    const int mb = (lane < 16) ? 0 : 8;
    if (n < H) {
        #pragma unroll
        for (int j = 0; j < 8; ++j) {
            long long r = m0 + mb + j;
            G[r * H + n] = c[j] + bih_s[n];
        }
    }
}

// =====================================================================
// Kernel 3: sequential RNN scan (tiny: 12x12 per step) + t_p head,
//           then angles x_p[B,36] = sigmoid(concat(a,v,t)) * pi
// =====================================================================
__global__ void rnn_scan_kernel(const float* __restrict__ G,
                                const float* __restrict__ W_hh, const float* __restrict__ b_hh,
                                const float* __restrict__ W_t,  const float* __restrict__ b_t,
                                const float* __restrict__ z_av, float* __restrict__ xp) {
    __shared__ float Whh[H * H], Wt[H * H], bhh[H], bt[H];
    int tid = threadIdx.x;
    for (int i = tid; i < H * H; i += blockDim.x) { Whh[i] = W_hh[i]; Wt[i] = W_t[i]; }
    if (tid < H) { bhh[tid] = b_hh[tid]; bt[tid] = b_t[tid]; }
    __syncthreads();

    int b = blockIdx.x * blockDim.x + tid;
    if (b >= B_BATCH) return;

    float h[H];
    #pragma unroll
    for (int i = 0; i < H; ++i) h[i] = 0.0f;

    const float* g = G + (long long)b * T_STEPS * H;
    for (int t = 0; t < T_STEPS; ++t) {
        float nh[H];
        #pragma unroll
        for (int i = 0; i < H; ++i) {
            float acc = g[t * H + i] + bhh[i];
            #pragma unroll
            for (int j = 0; j < H; ++j) acc += Whh[i * H + j] * h[j];
            nh[i] = tanhf(acc);
        }
        #pragma unroll
        for (int i = 0; i < H; ++i) h[i] = nh[i];
    }

    float* xpb = xp + (long long)b * 36;
    #pragma unroll
    for (int i = 0; i < 24; ++i) {
        float z = z_av[(long long)b * 24 + i];
        xpb[i] = PI_F / (1.0f + __expf(-z));
    }
    #pragma unroll
    for (int i = 0; i < H; ++i) {
        float z = bt[i];
        #pragma unroll
        for (int j = 0; j < H; ++j) z += Wt[i * H + j] * h[j];
        xpb[24 + i] = PI_F / (1.0f + __expf(-z));
    }
}

// =====================================================================
// Kernel 4: 12-qubit state-vector sim, entire state (32 KB) in LDS.
//           One workgroup (256 thr = 8 waves) per batch element.
// =====================================================================
__device__ __forceinline__ void apply1q(float* sRe, float* sIm, int wire,
                                        float u00r, float u00i, float u01r, float u01i,
                                        float u10r, float u10i, float u11r, float u11i,
                                        int tid) {
    const int m = 1 << (NQ - 1 - wire);
    for (int p = tid; p < DIM / 2; p += 256) {
        int i0 = ((p & ~(m - 1)) << 1) | (p & (m - 1));
        int i1 = i0 | m;
        float ar = sRe[i0], ai = sIm[i0];
        float br = sRe[i1], bi = sIm[i1];
        sRe[i0] = u00r * ar - u00i * ai + u01r * br - u01i * bi;
        sIm[i0] = u00r * ai + u00i * ar + u01r * bi + u01i * br;
        sRe[i1] = u10r * ar - u10i * ai + u11r * br - u11i * bi;
        sIm[i1] = u10r * ai + u10i * ar + u11r * bi + u11i * br;
    }
    __syncthreads();
}

__device__ __forceinline__ void applyXX(float* sRe, float* sIm, int wire,
                                        float c, float s, int tid) {
    const int m = 1 << (NQ - 2 - wire);          // i(j) = base + j*m, j = 2*bit_w + bit_{w+1}
    for (int p = tid; p < DIM / 4; p += 256) {
        int base = ((p & ~(m - 1)) << 2) | (p & (m - 1));
        int i0 = base, i1 = base + m, i2 = base + 2 * m, i3 = base + 3 * m;
        float r0 = sRe[i0], m0 = sIm[i0], r1 = sRe[i1], m1 = sIm[i1];
        float r2 = sRe[i2], m2 = sIm[i2], r3 = sRe[i3], m3 = sIm[i3];
        // new[j] = c*old[j] - i*s*old[3-j]
        sRe[i0] = c * r0 + s * m3;  sIm[i0] = c * m0 - s * r3;
        sRe[i1] = c * r1 + s * m2;  sIm[i1] = c * m1 - s * r2;
        sRe[i2] = c * r2 + s * m1;  sIm[i2] = c * m2 - s * r1;
        sRe[i3] = c * r3 + s * m0;  sIm[i3] = c * m3 - s * r0;
    }
    __syncthreads();
}

__device__ __forceinline__ void applyZZ(float* sRe, float* sIm, int wire,
                                        float c, float s, int tid) {
    const int bhi = NQ - 1 - wire, blo = bhi - 1;
    for (int k = tid; k < DIM; k += 256) {
        int diff = ((k >> bhi) ^ (k >> blo)) & 1;  // bits equal -> e=(c,-s); else conj(e)
        float ei = diff ? s : -s;
        float r = sRe[k], im = sIm[k];
        sRe[k] = r * c - im * ei;
        sIm[k] = r * ei + im * c;
    }
    __syncthreads();
}

__global__ void __launch_bounds__(256, 1)
quantum_kernel(const float* __restrict__ xp, const float* __restrict__ q_rot,
               const float* __restrict__ q_enta, const float* __restrict__ W_c,
               const float* __restrict__ b_c, float* __restrict__ out) {
    __shared__ float sRe[DIM];
    __shared__ float sIm[DIM];
    __shared__ float U[NQ * 8];        // per-qubit Rot matrices (reused every layer)
    __shared__ float red[8 * NQ];      // per-wave expval partials
    __shared__ float ev[NQ];

    const int tid = threadIdx.x;
    const int b   = blockIdx.x;

    for (int k = tid; k < DIM; k += 256) { sRe[k] = 0.0f; sIm[k] = 0.0f; }
    if (tid == 0) sRe[0] = 1.0f;

    if (tid < NQ) {
        const float* a = xp + (long long)b * 36 + tid * 3;
        float phi = a[0], th = a[1], om = a[2];
        float s, c;  __sincosf(0.5f * th, &s, &c);
        float sa, ca; __sincosf(0.5f * (phi + om), &sa, &ca);   // ep = (ca, -sa)
        float sb, cb; __sincosf(0.5f * (phi - om), &sb, &cb);   // em = (cb, -sb)
        float* u = U + tid * 8;
        u[0] =  ca * c;  u[1] = -sa * c;     // u00 = ep*c
        u[2] = -cb * s;  u[3] = -sb * s;     // u01 = -conj(em)*s
        u[4] =  cb * s;  u[5] = -sb * s;     // u10 = em*s
        u[6] =  ca * c;  u[7] =  sa * c;     // u11 = conj(ep)*c
    }
    __syncthreads();

    for (int layer = 0; layer < LREP; ++layer) {
        for (int i = 0; i < NQ; ++i) {
            const float* u = U + i * 8;
            apply1q(sRe, sIm, i, u[0], u[1], u[2], u[3], u[4], u[5], u[6], u[7], tid);
        }
        for (int j = 0; j < NQ - 1; ++j) {
            float wr = q_rot[layer * NQ + j];
            float s1, c1; __sincosf(0.5f * wr, &s1, &c1);
            if ((j & 1) == 0)   // Rx
                apply1q(sRe, sIm, j, c1, 0.f, 0.f, -s1, 0.f, -s1, c1, 0.f, tid);
            else                // Ry
                apply1q(sRe, sIm, j, c1, 0.f, -s1, 0.f, s1, 0.f, c1, 0.f, tid);

            float we = q_enta[layer * (NQ - 1) + j];
            float s2, c2; __sincosf(0.5f * we, &s2, &c2);
            if ((j & 1) == 0) applyXX(sRe, sIm, j, c2, s2, tid);
            else              applyZZ(sRe, sIm, j, c2, s2, tid);
        }
        {   // final 1q gate on wire NQ-1 = 11 (odd -> Ry)
            float wr = q_rot[layer * NQ + (NQ - 1)];
            float s1, c1; __sincosf(0.5f * wr, &s1, &c1);
            apply1q(sRe, sIm, NQ - 1, c1, 0.f, -s1, 0.f, s1, 0.f, c1, 0.f, tid);
        }
    }

    // PauliZ expvals: per-thread partials, wave32 shuffle reduce, cross-wave via LDS
    float part[NQ];
    #pragma unroll
    for (int q = 0; q < NQ; ++q) part[q] = 0.0f;
    for (int k = tid; k < DIM; k += 256) {
        float p = sRe[k] * sRe[k] + sIm[k] * sIm[k];
        #pragma unroll
        for (int q = 0; q < NQ; ++q)
            part[q] += ((k >> (NQ - 1 - q)) & 1) ? -p : p;
    }
    #pragma unroll
    for (int q = 0; q < NQ; ++q) {
        float v = part[q];
        #pragma unroll
        for (int off = 16; off > 0; off >>= 1) v += __shfl_xor(v, off, 32);
        if ((tid & 31) == 0) red[(tid >> 5) * NQ + q] = v;
    }
    __syncthreads();
    if (tid < NQ) {
        float e = 0.0f;
        #pragma unroll
        for (int w = 0; w < 8; ++w) e += red[w * NQ + tid];
        ev[tid] = e;
    }
    __syncthreads();
    if (tid < 2) {
        float o = b_c[tid];
        #pragma unroll
        for (int q = 0; q < NQ; ++q) o += W_c[tid * NQ + q] * ev[q];
        out[(long long)b * 2 + tid] = o;
    }
}

// =====================================================================
// launch
// =====================================================================
extern "C" void kernel_launch(void* const* d_in, const int* in_sizes, int n_in,
                              void* d_out, int out_size, void* d_ws, size_t ws_size,
                              hipStream_t stream) {
    (void)in_sizes; (void)n_in; (void)out_size; (void)ws_size;
    const float* x_a   = (const float*)d_in[0];
    const float* x_v   = (const float*)d_in[1];
    const float* x_t   = (const float*)d_in[2];
    const float* W_a   = (const float*)d_in[3];
    const float* b_a   = (const float*)d_in[4];
    const float* W_v   = (const float*)d_in[5];
    const float* b_v   = (const float*)d_in[6];
    const float* W_ih  = (const float*)d_in[7];
    const float* W_hh  = (const float*)d_in[8];
    const float* b_ih  = (const float*)d_in[9];
    const float* b_hh  = (const float*)d_in[10];
    const float* W_t   = (const float*)d_in[11];
    const float* b_t   = (const float*)d_in[12];
    const float* q_rot = (const float*)d_in[13];
    const float* q_enta= (const float*)d_in[14];
    const float* W_c   = (const float*)d_in[15];
    const float* b_c   = (const float*)d_in[16];
    float* out = (float*)d_out;

    // workspace layout (floats): G[102400*12] | z_av[B*24] | xp[B*36]  (~5.4 MB)
    float* G    = (float*)d_ws;
    float* z_av = G + (size_t)B_BATCH * T_STEPS * H;
    float* xp   = z_av + (size_t)B_BATCH * 24;

    proj_av_kernel<<<(B_BATCH * 24 + 255) / 256, 256, 0, stream>>>(
        x_a, x_v, W_a, b_a, W_v, b_v, z_av);

    rnn_in_gemm_kernel<<<(B_BATCH * T_STEPS) / (16 * GEMM_WAVES), 32 * GEMM_WAVES, 0, stream>>>(
        x_t, W_ih, b_ih, G);

    rnn_scan_kernel<<<(B_BATCH + 255) / 256, 256, 0, stream>>>(
        G, W_hh, b_hh, W_t, b_t, z_av, xp);

    quantum_kernel<<<B_BATCH, 256, 0, stream>>>(
        xp, q_rot, q_enta, W_c, b_c, out);
}